// Laplacian_3178275799590
// MI455X (gfx1250) — compile-verified
//
#include <hip/hip_runtime.h>
#include <math.h>

typedef __attribute__((ext_vector_type(2))) float v2f;
typedef __attribute__((ext_vector_type(4))) float v4f;
typedef __attribute__((ext_vector_type(8))) float v8f;

#define KSLICES     16
#define FILL_W      1.0f

// ---------------------------------------------------------------------------
// Pass 1: row sums via V_WMMA_F32_16X16X4_F32 with B = all-ones.
// D = A(16x4) x 1(4x16) + C  =>  every column of D holds the 16 row sums.
// Wave W handles row-tile (W / KSLICES), K-slice (W % KSLICES).
// A layout (32-bit A 16x4): lane = M (lane&15); lanes 0-15 carry K=0,1,
// lanes 16-31 carry K=2,3 in the two A VGPRs -> one b64 load per fragment.
//
// Pipelining: 16 fragments (64 K-values) are loaded into distinct registers
// per batch so the wave keeps 16 loads in flight, then 16 WMMAs are issued
// alternating between two accumulators to break the D->C dependency chain.
// ---------------------------------------------------------------------------
__global__ void lap_rowsum_wmma(const float* __restrict__ adj,
                                float* __restrict__ partial, int N) {
    const int lane  = threadIdx.x & 31;
    const int wave  = threadIdx.x >> 5;
    const int W     = blockIdx.x * (blockDim.x >> 5) + wave;
    const int rowTile = W >> 4;                 // / KSLICES
    const int kSlice  = W & (KSLICES - 1);
    const int kLen    = N / KSLICES;            // 512 for N=8192

    const int m    = lane & 15;
    const int kOff = (lane >> 4) << 1;          // 0 or 2
    const size_t row = (size_t)(rowTile * 16 + m);
    const float* p = adj + row * (size_t)N + (size_t)kSlice * kLen + kOff;

    v8f c0 = {};
    v8f c1 = {};
    v2f b;  b.x = 1.0f;  b.y = 1.0f;

    for (int k = 0; k < kLen; k += 64) {        // 16 WMMA steps per batch
        v2f a[16];
#pragma unroll
        for (int t = 0; t < 16; ++t)            // 16 b64 loads in flight
            a[t] = *(const v2f*)(p + k + 4 * t);

        __builtin_prefetch(p + k + 2048, 0, 3); // WGP-scope prefetch, 8KB ahead

#pragma unroll
        for (int t = 0; t < 16; t += 2) {
            c0 = __builtin_amdgcn_wmma_f32_16x16x4_f32(
                     false, a[t + 0], false, b, (short)0, c0, false, false);
            c1 = __builtin_amdgcn_wmma_f32_16x16x4_f32(
                     false, a[t + 1], false, b, (short)0, c1, false, false);
        }
    }

    const v8f c = c0 + c1;                      // fixed-order combine

    // lane 0 owns rows rowTile*16+0..7, lane 16 owns rows +8..15 (N=0 column).
    if ((lane & 15) == 0) {
        const int rb = rowTile * 16 + ((lane >> 4) << 3);
        float* q = partial + (size_t)kSlice * N + rb;
        q[0] = c[0]; q[1] = c[1]; q[2] = c[2]; q[3] = c[3];
        q[4] = c[4]; q[5] = c[5]; q[6] = c[6]; q[7] = c[7];
    }
}

// ---------------------------------------------------------------------------
// Pass 1b: deterministic fixed-order reduction of the K-slice partials,
// fold in FILL_W (the +I adds exactly 1.0 to every row sum), dp = d^-0.5.
// ---------------------------------------------------------------------------
__global__ void lap_finalize_dp(const float* __restrict__ partial,
                                float* __restrict__ dp, int N) {
    const int i = blockIdx.x * blockDim.x + threadIdx.x;
    if (i >= N) return;
    float s = FILL_W;
#pragma unroll
    for (int t = 0; t < KSLICES; ++t) s += partial[(size_t)t * N + i];
    dp[i] = 1.0f / sqrtf(s);
}

// ---------------------------------------------------------------------------
// Pass 2: out[i,j] = dp[i]*dp[j]*(A[i,j] + [i==j]).  Pure stream:
// b128 loads, non-temporal b128 stores so the 256MB write stream does not
// evict A lines still resident in the 192MB L2 from pass 1.
// ---------------------------------------------------------------------------
__global__ void lap_scale(const float* __restrict__ adj,
                          const float* __restrict__ dp,
                          float* __restrict__ out, int N) {
    const unsigned nq   = (unsigned)(N >> 2);
    const unsigned flat = blockIdx.x * blockDim.x + threadIdx.x;  // over N*N/4
    const unsigned row  = flat / nq;
    const unsigned c4   = flat - row * nq;

    v4f a  = *((const v4f*)adj + flat);
    v4f dj = *((const v4f*)dp + c4);
    const float di = dp[row];

    const unsigned j0 = c4 << 2;
    const unsigned d  = row - j0;          // unsigned wrap: in-range iff < 4
    if (d < 4u) {
        if      (d == 0u) a.x += FILL_W;
        else if (d == 1u) a.y += FILL_W;
        else if (d == 2u) a.z += FILL_W;
        else              a.w += FILL_W;
    }

    v4f r;
    r.x = di * dj.x * a.x;
    r.y = di * dj.y * a.y;
    r.z = di * dj.z * a.z;
    r.w = di * dj.w * a.w;
    __builtin_nontemporal_store(r, (v4f*)out + flat);   // NT b128 store
}

// ---------------------------------------------------------------------------
extern "C" void kernel_launch(void* const* d_in, const int* in_sizes, int n_in,
                              void* d_out, int out_size, void* d_ws, size_t ws_size,
                              hipStream_t stream) {
    const float* adj = (const float*)d_in[0];
    float*       out = (float*)d_out;

    const long long tot = (long long)in_sizes[0];
    const int N = (int)llround(sqrt((double)tot));      // 8192

    float* partial = (float*)d_ws;                      // KSLICES * N floats
    float* dp      = partial + (size_t)KSLICES * N;     // N floats, 16B aligned

    // Pass 1: (N/16) row tiles x KSLICES k-slices, 8 waves (256 thr) per block.
    const int rowTiles = N / 16;
    const int waves    = rowTiles * KSLICES;
    const int blocks1  = waves / 8;
    lap_rowsum_wmma<<<blocks1, 256, 0, stream>>>(adj, partial, N);

    // Pass 1b: dp[i] = (rowsum + 1)^-1/2
    lap_finalize_dp<<<(N + 255) / 256, 256, 0, stream>>>(partial, dp, N);

    // Pass 2: one float4 per thread over N*N/4 elements.
    const unsigned total4 = (unsigned)((long long)N * N / 4);
    lap_scale<<<total4 / 256, 256, 0, stream>>>(adj, dp, out, N);
}